// MakeCutouts_1090921693623
// MI455X (gfx1250) — compile-verified
//
#include <hip/hip_runtime.h>

#define HH   1024
#define WW   1024
#define CH   3
#define IW   1025      // padded integral-image dimension
#define CS   224
#define NCUT 128

typedef __attribute__((ext_vector_type(2))) float        v2f;
typedef __attribute__((ext_vector_type(8))) float        v8f;
typedef __attribute__((ext_vector_type(4))) unsigned int u32x4;
typedef __attribute__((ext_vector_type(8))) unsigned int u32x8;

// LDS row stride in DWORDs after TDM padding: 512 data + 2 pads of 2 DWORDs
#define LROW 516

// ---------------------------------------------------------------------------
// TDM: DMA a 16-row x 512-col f32 tile (row stride 1024 elems) into LDS with
// 2-DWORD padding every 256 DWORDs (bank-conflict-free for the WMMA A reads).
// D# layout per CDNA5 ISA; 2D tensor -> groups 2/3 NULL.
// ---------------------------------------------------------------------------
__device__ __forceinline__ void tdm_load_half(unsigned ldsByte,
                                              unsigned long long gaddr) {
    u32x4 g0;
    g0[0] = 0x1u;                                  // count=1, user descriptor
    g0[1] = ldsByte;                               // lds_addr (bytes)
    g0[2] = (unsigned)gaddr;                       // global_addr[31:0]
    g0[3] = (unsigned)(gaddr >> 32) | 0x80000000u; // global_addr[56:32] | type=2
    u32x8 g1;
    g1[0] = (2u << 16)                             // data_size = 4 bytes
          | (1u << 20)                             // pad_enable
          | (7u << 22)                             // pad_interval: 256 DWORDs
          | (1u << 25);                            // pad_amount: 2 DWORDs
    g1[1] = (1024u & 0xffffu) << 16;               // tensor_dim0 = 1024 (lo16)
    g1[2] = (1024u & 0xffffu) << 16;               // td0 hi=0 | tensor_dim1 lo16
    g1[3] = (512u << 16);                          // td1 hi=0 | tile_dim0 = 512
    g1[4] = 16u;                                   // tile_dim1 = 16, tile_dim2 = 0
    g1[5] = 1024u;                                 // tensor_dim0_stride = 1024
    g1[6] = 0u;
    g1[7] = 0u;
    asm volatile("tensor_load_to_lds %0, %1" :: "s"(g0), "s"(g1) : "memory");
}

// ---------------------------------------------------------------------------
// Kernel 1: row-wise inclusive prefix sum via V_WMMA_F32_16X16X4_F32.
// Matrix-scan: rowprefix(X) = X @ U (U = upper-triangular ones), K chunked 4x4.
// One wave per 16-row strip; TDM double-issue stages the strip into LDS and
// overlaps the second half's DMA with the first half's WMMA tiles.
// ---------------------------------------------------------------------------
__global__ __launch_bounds__(32) void row_scan_wmma(const float* __restrict__ x,
                                                    float* __restrict__ P) {
    __shared__ float tile[2 * 16 * LROW];          // 2 halves, 64.5 KB total
    const int strip = blockIdx.x;                  // 0..191 (3 ch * 64 strips)
    const int c     = strip >> 6;
    const int row0  = (strip & 63) << 4;
    const int lane  = threadIdx.x;                 // wave32
    const int half  = lane >> 4;                   // lane half: 0 or 1
    const int l16   = lane & 15;

    const float* src = x + ((size_t)c * HH + row0) * WW;
    float* dstBase   = P + (size_t)c * IW * IW + (size_t)(row0 + 1) * IW + 1;

    // issue both strip-half DMAs up front (in-order, tracked by TENSORcnt)
    const unsigned ldsBase = (unsigned)(unsigned long long)&tile[0];
    tdm_load_half(ldsBase,                    (unsigned long long)src);
    tdm_load_half(ldsBase + 16u * LROW * 4u,  (unsigned long long)(src + 512));

    float carry[8];
#pragma unroll
    for (int e = 0; e < 8; ++e) carry[e] = 0.0f;

    for (int hb = 0; hb < 2; ++hb) {
        if (hb == 0) asm volatile("s_wait_tensorcnt 0x1" ::: "memory");
        else         asm volatile("s_wait_tensorcnt 0x0" ::: "memory");
        const float* lhalf = &tile[hb * 16 * LROW];

        for (int t = 0; t < 32; ++t) {
            const int kbase = t << 4;              // tile col base within half

            // D = X @ U accumulated over 4 K-chunks of 4
            v8f acc = {};
#pragma unroll
            for (int kc = 0; kc < 4; ++kc) {
                const int k0 = (kc << 2) + (half << 1); // K pos within tile
                const int kl = kbase + k0;              // K pos within half
                const int li = l16 * LROW + kl + ((kl >> 8) << 1);
                v2f a, b;
                a.x = lhalf[li];                        // A: M=l16, K=k0
                a.y = lhalf[li + 1];                    //    M=l16, K=k0+1
                b.x = (k0     <= l16) ? 1.0f : 0.0f;    // B: U[k0,   N=l16]
                b.y = (k0 + 1 <= l16) ? 1.0f : 0.0f;    //    U[k0+1, N=l16]
                acc = __builtin_amdgcn_wmma_f32_16x16x4_f32(
                          false, a, false, b, (short)0, acc, false, false);
            }

            // add running carry; new carry = column 15 of each half-wave (DPP)
            float d[8];
#pragma unroll
            for (int e = 0; e < 8; ++e) {
                d[e]     = acc[e] + carry[e];
                carry[e] = __shfl(d[e], 15, 16);
            }
            // store: row m = e + 8*half, col n = l16 (C/D VGPR layout)
            const int x0 = (hb << 9) + kbase;
#pragma unroll
            for (int e = 0; e < 8; ++e)
                dstBase[(size_t)(e + 8 * half) * IW + x0 + l16] = d[e];
        }
    }
}

// ---------------------------------------------------------------------------
// Kernel 2: in-place column cumsum of P + pad row/col zeros.
// Thread-per-column: loads/stores at fixed y are unit-stride -> coalesced.
// ---------------------------------------------------------------------------
__global__ void col_scan(float* __restrict__ P) {
    const int g = blockIdx.x * blockDim.x + threadIdx.x;
    if (g >= CH * IW) return;
    const int c  = g / IW;
    const int xx = g - c * IW;
    float* base = P + (size_t)c * IW * IW;
    base[xx] = 0.0f;                          // y = 0 pad row
    if (xx == 0) {                            // x = 0 pad column
        for (int y = 1; y < IW; ++y) base[(size_t)y * IW] = 0.0f;
    } else {
        float sum = 0.0f;
        for (int y = 1; y < IW; ++y) {
            const size_t idx = (size_t)y * IW + xx;
            sum += base[idx];
            base[idx] = sum;
        }
    }
}

// ---------------------------------------------------------------------------
// Kernel 3: 4-corner SAT gather per output pixel (L2-resident reads).
// out[n][c][i][j], n outermost — matches reference transpose/reshape order.
// ---------------------------------------------------------------------------
__global__ void cutout_gather(const float* __restrict__ I,
                              const int* __restrict__ sizes,
                              const int* __restrict__ oy,
                              const int* __restrict__ ox,
                              float* __restrict__ out, int total) {
    const int g = blockIdx.x * blockDim.x + threadIdx.x;
    if (g >= total) return;
    const int j = g % CS;
    const int i = (g / CS) % CS;
    const int c = (g / (CS * CS)) % CH;
    const int n = g / (CS * CS * CH);

    const int s    = sizes[n];
    const int lo_i = (i * s) / CS;
    const int hi_i = ((i + 1) * s + (CS - 1)) / CS;   // ceil
    const int lo_j = (j * s) / CS;
    const int hi_j = ((j + 1) * s + (CS - 1)) / CS;

    const int r0 = oy[n] + lo_i, r1 = oy[n] + hi_i;
    const int c0 = ox[n] + lo_j, c1 = ox[n] + hi_j;

    const float* Ic = I + (size_t)c * IW * IW;
    const float S = Ic[(size_t)r1 * IW + c1] - Ic[(size_t)r0 * IW + c1]
                  - Ic[(size_t)r1 * IW + c0] + Ic[(size_t)r0 * IW + c0];
    const float area = (float)((hi_i - lo_i) * (hi_j - lo_j));
    out[g] = S / area;
}

extern "C" void kernel_launch(void* const* d_in, const int* in_sizes, int n_in,
                              void* d_out, int out_size, void* d_ws, size_t ws_size,
                              hipStream_t stream) {
    const float* xin   = (const float*)d_in[0];
    const int*   sizes = (const int*)d_in[1];
    const int*   oy    = (const int*)d_in[2];
    const int*   ox    = (const int*)d_in[3];
    float*       out   = (float*)d_out;
    float*       P     = (float*)d_ws;       // 3*1025*1025 f32 = 12.6 MB scratch

    row_scan_wmma<<<CH * (HH / 16), 32, 0, stream>>>(xin, P);
    col_scan<<<(CH * IW + 255) / 256, 256, 0, stream>>>(P);
    const int total = NCUT * CH * CS * CS;
    cutout_gather<<<(total + 255) / 256, 256, 0, stream>>>(P, sizes, oy, ox, out, total);
}